// KNN_29111288332315
// MI455X (gfx1250) — compile-verified
//
#include <hip/hip_runtime.h>
#include <hip/hip_bf16.h>

// ---------------------------------------------------------------------------
// KNN tail: reference sorts dists over the QUERY axis and keeps the last
// `topk` entry-columns. So we only need D[n, q] = <x_q, table_n> for the last
// topk entries, then a full argsort of each 2048-long column.
// ---------------------------------------------------------------------------

typedef __attribute__((ext_vector_type(2))) float v2f;
typedef __attribute__((ext_vector_type(8))) float v8f;

#define DIM 512           // reference DIM (compile-time stride)
#define SORT_N 2048       // reference N_QUERIES (power of two -> bitonic)

// ---------------------------------------------------------------------------
// Kernel 1: f32 WMMA GEMM.  One wave per 16x16 output tile.
//   A = x           (M=queries, K=dim)    16x4 f32 fragment per wmma
//   B = table tail  (K=dim, N=entries)    4x16 f32 fragment per wmma
//   C tile -> scratch dists[n][q] (column-major over q for the sort kernel)
//
// 16x4 f32 A layout (ISA 7.12.2): lanes 0-15 hold M=lane, VGPR0/1 = K+0,K+1;
// lanes 16-31 hold M=lane-16, VGPR0/1 = K+2,K+3.  B mirrors with N across
// lanes.  Both fragments are therefore contiguous float2 (b64) loads.
// ---------------------------------------------------------------------------
__global__ __launch_bounds__(256) void knn_wmma_gemm(
    const float* __restrict__ x,      // [nq, DIM]
    const float* __restrict__ tbl,    // [ne, DIM]
    float* __restrict__ dists,        // [topk, nq]  (scratch, column-major in q)
    int nq, int n0, int tiles_n)
{
    const int lane   = threadIdx.x & 31;
    const int wave   = threadIdx.x >> 5;
    const int wid    = blockIdx.x * (blockDim.x >> 5) + wave;
    const int m_tile = wid / tiles_n;          // 0..nq/16-1
    const int n_tile = wid % tiles_n;          // 0..tiles_n-1

    const int row  = lane & 15;                // M (for A) / N (for B)
    const int koff = (lane >> 4) << 1;         // 0 or 2: K sub-offset per half-wave

    const float* aPtr = x   + (size_t)(m_tile * 16 + row)      * DIM + koff;
    const float* bPtr = tbl + (size_t)(n0 + n_tile * 16 + row) * DIM + koff;

    v8f c = {};
#pragma unroll 8
    for (int k = 0; k < DIM; k += 4) {
        v2f a = *(const v2f*)(aPtr + k);       // global_load_b64
        v2f b = *(const v2f*)(bPtr + k);       // global_load_b64
        // D = A*B + C, f32 in / f32 accum (matches the f32 reference exactly)
        c = __builtin_amdgcn_wmma_f32_16x16x4_f32(
                /*neg_a=*/false, a, /*neg_b=*/false, b,
                /*c_mod=*/(short)0, c, /*reuse_a=*/false, /*reuse_b=*/false);
    }

    // C/D layout: VGPR v, lanes 0-15 -> M=v, lanes 16-31 -> M=v+8; N = lane&15
    const int q0 = m_tile * 16 + ((lane >> 4) << 3);
    const int n  = n_tile * 16 + row;
    float* dcol  = dists + (size_t)n * nq;
#pragma unroll
    for (int v = 0; v < 8; ++v)
        dcol[q0 + v] = c[v];
}

// ---------------------------------------------------------------------------
// Kernel 2: per-column bitonic argsort in LDS (ascending; ties broken by
// original index to match a stable argsort).  One block per entry column.
// 2048 * (4B val + 4B idx) = 16 KB LDS, well under the 320 KB/WGP budget.
// ---------------------------------------------------------------------------
__global__ __launch_bounds__(1024) void knn_col_argsort(
    const float* __restrict__ dists,   // [topk, nq]
    int*   __restrict__ out_idx,       // [nq, topk]
    float* __restrict__ out_val,       // [nq, topk]
    int nq, int topk)
{
    __shared__ float sv[SORT_N];
    __shared__ int   si[SORT_N];

    const int col = blockIdx.x;                    // 0..topk-1
    const int t   = threadIdx.x;
    const float* src = dists + (size_t)col * nq;

    for (int i = t; i < nq; i += blockDim.x) {
        sv[i] = src[i];
        si[i] = i;
    }
    __syncthreads();

    for (int k = 2; k <= nq; k <<= 1) {
        for (int s = k >> 1; s > 0; s >>= 1) {
            for (int i = t; i < nq; i += blockDim.x) {
                const int partner = i ^ s;
                if (partner > i) {
                    const bool up = ((i & k) == 0);        // ascending sub-block
                    float v0 = sv[i], v1 = sv[partner];
                    int   i0 = si[i], i1 = si[partner];
                    const bool gt = (v0 > v1) || (v0 == v1 && i0 > i1);
                    if (gt == up) {
                        sv[i] = v1; sv[partner] = v0;
                        si[i] = i1; si[partner] = i0;
                    }
                }
            }
            __syncthreads();
        }
    }

    // top_indices then top_dists, each row-major (nq, topk)
    for (int i = t; i < nq; i += blockDim.x) {
        out_val[(size_t)i * topk + col] = sv[i];
        out_idx[(size_t)i * topk + col] = si[i];
    }
}

// ---------------------------------------------------------------------------
extern "C" void kernel_launch(void* const* d_in, const int* in_sizes, int n_in,
                              void* d_out, int out_size, void* d_ws, size_t ws_size,
                              hipStream_t stream)
{
    const float* x   = (const float*)d_in[0];   // [nq, DIM] f32
    const float* tbl = (const float*)d_in[1];   // [ne, DIM] f32

    const int nq   = in_sizes[0] / DIM;          // 2048
    const int ne   = in_sizes[1] / DIM;          // 131072
    const int topk = out_size / (2 * nq);        // 32 (indices + dists concatenated)
    const int n0   = ne - topk;                  // first entry column we need

    float* dists = (float*)d_ws;                 // topk * nq f32 = 256 KB scratch

    // GEMM: one wave per 16x16 tile, 8 waves per 256-thread block
    const int tiles_m = nq / 16;                 // 128
    const int tiles_n = topk / 16;               // 2
    const int waves   = tiles_m * tiles_n;       // 256
    const int wpb     = 8;
    knn_wmma_gemm<<<dim3((waves + wpb - 1) / wpb), dim3(32 * wpb), 0, stream>>>(
        x, tbl, dists, nq, n0, tiles_n);

    // argsort each column; outputs: [nq*topk] int32 indices, then [nq*topk] f32
    int*   out_idx = (int*)d_out;
    float* out_val = (float*)d_out + (size_t)nq * topk;
    knn_col_argsort<<<dim3(topk), dim3(1024), 0, stream>>>(
        dists, out_idx, out_val, nq, topk);
}